// CITADELEncoder_26551487824589
// MI455X (gfx1250) — compile-verified
//
#include <hip/hip_runtime.h>
#include <hip/hip_bf16.h>

typedef __attribute__((ext_vector_type(2))) float v2f;
typedef __attribute__((ext_vector_type(8))) float v8f;

#define BB 16
#define SS 257
#define SP 256
#define HH 768
#define VV 30522
#define PP 32
#define TOPK 8

// Output layout (floats), concatenated in reference return order.
#define O_COND   ((size_t)0)                          // (1,1)
#define O_MARG   ((size_t)1)                          // (1,1)
#define O_RMSEQ  ((size_t)2)                          // (16,30522)
#define O_SMSUM  (O_RMSEQ + (size_t)BB*VV)            // (16,30522)
#define O_MASK   (O_SMSUM + (size_t)BB*VV)            // (16,256)
#define O_RREPR  (O_MASK  + (size_t)BB*SP)            // (16,30522)
#define O_EIDS   (O_RREPR + (size_t)BB*VV)            // (16,256,8)
#define O_EREPR  (O_EIDS  + (size_t)BB*SP*TOPK)       // (16,256,32)
#define O_EW     (O_EREPR + (size_t)BB*SP*PP)         // (16,256,8)

// ---------------------------------------------------------------------------
// Pass 1: one block per (b,s) row (mask is constant across the row).
// Online softmax stats + top-8 ranked by the monotone key relu(x)
// (key forced to 0 if row mask == 0, so tie ordering matches log1p(relu)*mk).
// log1p is applied only to the 8 winners at the end.
// ---------------------------------------------------------------------------
__global__ __launch_bounds__(256)
void k_row_pass(const float* __restrict__ logits,
                const float* __restrict__ amask,
                float* __restrict__ out,
                float* __restrict__ rowmax,
                float* __restrict__ rowsum)
{
    const int row = blockIdx.x;            // 0..4095
    const int b   = row >> 8;
    const int s   = row & 255;
    const int tid = threadIdx.x;
    const float mk = amask[b * SS + s + 1];
    const bool  live = (mk > 0.f);
    const float2* x2 = reinterpret_cast<const float2*>(
        logits + ((size_t)(b * SS + s + 1)) * VV);   // row base is 8B aligned

    float m = -3.4e38f, sum = 0.f;
    float tv[TOPK]; int ti[TOPK];
#pragma unroll
    for (int j = 0; j < TOPK; ++j) { tv[j] = -1.f; ti[j] = 0x7fffffff; }

    for (int p = tid; p < VV / 2; p += 256) {       // VV even: exact pairs
        float2 xp = x2[p];
#pragma unroll
        for (int e = 0; e < 2; ++e) {
            float xv = (e == 0) ? xp.x : xp.y;
            int   v  = 2 * p + e;
            // branchy online softmax: common path costs one v_exp
            if (xv > m) { sum = sum * __expf(m - xv) + 1.0f; m = xv; }
            else        { sum += __expf(xv - m); }
            // selection key: relu(x) for live rows, 0 for masked rows
            float key = live ? fmaxf(xv, 0.f) : 0.f;
            if (key > tv[TOPK - 1] ||
                (key == tv[TOPK - 1] && v < ti[TOPK - 1])) {
                tv[TOPK - 1] = key; ti[TOPK - 1] = v;
#pragma unroll
                for (int j = TOPK - 1; j > 0; --j) {
                    bool sw = (tv[j] > tv[j - 1]) ||
                              (tv[j] == tv[j - 1] && ti[j] < ti[j - 1]);
                    if (sw) {
                        float fv = tv[j]; tv[j] = tv[j - 1]; tv[j - 1] = fv;
                        int   ii = ti[j]; ti[j] = ti[j - 1]; ti[j - 1] = ii;
                    }
                }
            }
        }
    }

    __shared__ float s_a[256];
    __shared__ float s_b[256];
    __shared__ int   s_i[256];

    // Combine (max, sum) online-softmax state across the block.
    s_a[tid] = m; s_b[tid] = sum;
    __syncthreads();
    for (int off = 128; off > 0; off >>= 1) {
        if (tid < off) {
            float m2 = s_a[tid + off], s2 = s_b[tid + off];
            float m1 = s_a[tid],       s1 = s_b[tid];
            float nm = fmaxf(m1, m2);
            s_b[tid] = s1 * __expf(m1 - nm) + s2 * __expf(m2 - nm);
            s_a[tid] = nm;
        }
        __syncthreads();
    }
    if (tid == 0) {
        rowmax[row] = s_a[0];
        rowsum[row] = s_b[0];
        out[O_MASK + row] = mk;
    }
    __syncthreads();

    // 8 rounds: block-wide argmax of each thread's current head, pop winner.
    for (int k = 0; k < TOPK; ++k) {
        s_a[tid] = tv[0]; s_i[tid] = ti[0];
        __syncthreads();
        for (int off = 128; off > 0; off >>= 1) {
            if (tid < off) {
                float v2 = s_a[tid + off]; int i2 = s_i[tid + off];
                if (v2 > s_a[tid] || (v2 == s_a[tid] && i2 < s_i[tid])) {
                    s_a[tid] = v2; s_i[tid] = i2;
                }
            }
            __syncthreads();
        }
        float wv = s_a[0]; int wi = s_i[0];
        if (tid == 0) {
            // weight = mk * log1p(relu(x)); key already is relu(x) (or 0)
            float w = (wv > 0.f) ? mk * log1pf(wv) : 0.f;
            out[O_EW   + (size_t)row * TOPK + k] = w;
            out[O_EIDS + (size_t)row * TOPK + k] = (float)wi;  // exact: V < 2^24
        }
        if (tv[0] == wv && ti[0] == wi) {   // vocab index unique -> one winner
#pragma unroll
            for (int j = 0; j < TOPK - 1; ++j) { tv[j] = tv[j + 1]; ti[j] = ti[j + 1]; }
            tv[TOPK - 1] = -1.f; ti[TOPK - 1] = 0x7fffffff;
        }
        __syncthreads();
    }
}

// ---------------------------------------------------------------------------
// Pass 2: each thread owns two consecutive v (float2 loads); loop over s.
// exp(x-m)/S folded to exp(x - (m + ln S)); router_repr via monotonicity:
// max_s log1p(relu(x)) = log1p(relu(max_s x)) over unmasked s.
// ---------------------------------------------------------------------------
__global__ __launch_bounds__(256)
void k_col_pass(const float* __restrict__ logits,
                const float* __restrict__ amask,
                const float* __restrict__ rowmax,
                const float* __restrict__ rowsum,
                float* __restrict__ out)
{
    const int chunks = (VV + 511) / 512;     // 60
    const int b     = blockIdx.x / chunks;
    const int chunk = blockIdx.x % chunks;
    const int tid   = threadIdx.x;
    const int v0    = chunk * 512 + tid * 2;

    __shared__ float s_c[SP];     // rowmax + ln(rowsum)
    __shared__ float s_bias[SP];  // 0 if mask>0 else -3.4e38 (excludes from max)
    {
        int r = b * SP + tid;
        s_c[tid]    = rowmax[r] + __logf(rowsum[r]);
        s_bias[tid] = (amask[b * SS + tid + 1] > 0.f) ? 0.f : -3.4e38f;
    }
    __syncthreads();
    if (v0 >= VV) return;

    const float2* base2 = reinterpret_cast<const float2*>(
        logits + ((size_t)(b * SS + 1)) * VV);      // (b,s0) base; 8B aligned
    const size_t vh = (size_t)(v0 >> 1);
    const size_t stride2 = (size_t)(VV / 2);

    float sm0 = 0.f, sm1 = 0.f;
    float M0 = -3.4e38f, M1 = -3.4e38f;
    for (int s = 0; s < SP; ++s) {
        if (s + 8 < SP)
            __builtin_prefetch(base2 + (size_t)(s + 8) * stride2 + vh, 0, 1);
        float2 xv = base2[(size_t)s * stride2 + vh];
        float c = s_c[s], bia = s_bias[s];
        sm0 += __expf(xv.x - c);
        sm1 += __expf(xv.y - c);
        M0 = fmaxf(M0, xv.x + bia);
        M1 = fmaxf(M1, xv.y + bia);
    }
    float2 rs; rs.x = sm0; rs.y = sm1;
    float2 rr;
    rr.x = (M0 > 0.f) ? log1pf(M0) : 0.f;
    rr.y = (M1 > 0.f) ? log1pf(M1) : 0.f;
    // O_SMSUM, O_RREPR are even offsets and b*VV+v0 is even -> 8B aligned
    *reinterpret_cast<float2*>(&out[O_SMSUM + (size_t)b * VV + v0]) = rs;
    *reinterpret_cast<float2*>(&out[O_RREPR + (size_t)b * VV + v0]) = rr;
}

// ---------------------------------------------------------------------------
// expert_repr GEMM via V_WMMA_F32_16X16X4_F32: one wave per 16-row M tile,
// full N=32 via two accumulators (A operand reused). K loop over 768 in 4s.
// ---------------------------------------------------------------------------
__global__ __launch_bounds__(32)
void k_expert_gemm(const float* __restrict__ hidden,
                   const float* __restrict__ W,
                   const float* __restrict__ bias,
                   const float* __restrict__ amask,
                   float* __restrict__ out)
{
    const int lane  = threadIdx.x;           // 0..31, full wave active
    const int m0    = blockIdx.x * 16;       // 256 blocks cover M=4096
    const int mr    = m0 + (lane & 15);      // A-matrix row for this lane
    const int khalf = lane >> 4;             // 0: K=k0,k0+1  1: K=k0+2,k0+3
    const int n     = lane & 15;

    // A row pointer: hiddens skips position 0 along S
    const float* Arow = hidden + ((size_t)((mr >> 8) * SS + (mr & 255) + 1)) * HH;
    // B(k,n) = W_tok[n][k] (W_tok is P x H row-major)
    const float* W0 = W + (size_t)n * HH;
    const float* W1 = W + (size_t)(16 + n) * HH;

    v8f c0 = {}; v8f c1 = {};
    for (int k0 = 0; k0 < HH; k0 += 4) {
        const int kk = k0 + khalf * 2;
        v2f a  = { Arow[kk], Arow[kk + 1] };
        v2f b0 = { W0[kk],   W0[kk + 1]   };
        v2f b1 = { W1[kk],   W1[kk + 1]   };
        c0 = __builtin_amdgcn_wmma_f32_16x16x4_f32(false, a, false, b0,
                                                   (short)0, c0, false, false);
        c1 = __builtin_amdgcn_wmma_f32_16x16x4_f32(false, a, false, b1,
                                                   (short)0, c1, false, false);
    }

    const float bi0 = bias[n], bi1 = bias[16 + n];
#pragma unroll
    for (int r = 0; r < 8; ++r) {
        int m = m0 + r + khalf * 8;          // C/D layout: lanes16-31 hold M=8..15
        float mk = amask[(m >> 8) * SS + (m & 255) + 1];
        out[O_EREPR + (size_t)m * PP + n]      = (c0[r] + bi0) * mk;
        out[O_EREPR + (size_t)m * PP + 16 + n] = (c1[r] + bi1) * mk;
    }
}

// ---------------------------------------------------------------------------
// Cleanup kernels: zero init, rm_seq scatter + avg_cond, avg_marg reduce.
// ---------------------------------------------------------------------------
__global__ __launch_bounds__(256)
void k_zero(float* __restrict__ out)
{
    size_t i = (size_t)blockIdx.x * 256 + threadIdx.x;
    if (i < (size_t)BB * VV) out[O_RMSEQ + i] = 0.f;
    if (i == 0) { out[O_COND] = 0.f; out[O_MARG] = 0.f; }
}

__global__ __launch_bounds__(256)
void k_scatter(float* __restrict__ out)
{
    int row = blockIdx.x * 256 + threadIdx.x;   // 0..4095
    if (row >= BB * SP) return;
    int b = row >> 8;
    int cnt = 0;
#pragma unroll
    for (int k = 0; k < TOPK; ++k) {
        float w = out[O_EW + (size_t)row * TOPK + k];
        if (w > 0.f) {
            int id = (int)out[O_EIDS + (size_t)row * TOPK + k];
            atomicAdd(&out[O_RMSEQ + (size_t)b * VV + id], 1.0f);
            ++cnt;
        }
    }
    // avg_cond = (1/B) * total number of positive expert weights
    atomicAdd(&out[O_COND], (float)cnt * (1.0f / BB));
}

__global__ __launch_bounds__(256)
void k_marg(float* __restrict__ out)
{
    __shared__ float sd[256];
    int v = blockIdx.x * 256 + threadIdx.x;
    float mx = 0.f;
    if (v < VV) {
#pragma unroll
        for (int b = 0; b < BB; ++b)
            mx = fmaxf(mx, out[O_RMSEQ + (size_t)b * VV + v]);
    }
    sd[threadIdx.x] = mx;
    __syncthreads();
    for (int off = 128; off > 0; off >>= 1) {
        if (threadIdx.x < off) sd[threadIdx.x] += sd[threadIdx.x + off];
        __syncthreads();
    }
    if (threadIdx.x == 0) atomicAdd(&out[O_MARG], sd[0]);
}

// ---------------------------------------------------------------------------
extern "C" void kernel_launch(void* const* d_in, const int* in_sizes, int n_in,
                              void* d_out, int out_size, void* d_ws, size_t ws_size,
                              hipStream_t stream)
{
    const float* hidden = (const float*)d_in[0];   // (16,257,768)
    const float* logits = (const float*)d_in[1];   // (16,257,30522)
    const float* amask  = (const float*)d_in[2];   // (16,257)
    const float* W_tok  = (const float*)d_in[3];   // (32,768)
    const float* b_tok  = (const float*)d_in[4];   // (32,)
    float* out = (float*)d_out;

    float* rowmax = (float*)d_ws;                  // 4096 floats
    float* rowsum = rowmax + BB * SP;              // 4096 floats

    // init rm_seq / scalars
    k_zero<<<((size_t)BB * VV + 255) / 256, 256, 0, stream>>>(out);

    // pass 1: row stats + top-8 + mask copy
    k_row_pass<<<BB * SP, 256, 0, stream>>>(logits, amask, out, rowmax, rowsum);

    // pass 2: router_softmax_sum + router_repr
    k_col_pass<<<BB * ((VV + 511) / 512), 256, 0, stream>>>(logits, amask,
                                                            rowmax, rowsum, out);

    // expert_repr GEMM (WMMA f32)
    k_expert_gemm<<<(BB * SP) / 16, 32, 0, stream>>>(hidden, W_tok, b_tok,
                                                     amask, out);

    // rm_seq scatter + avg_cond, then avg_marg
    k_scatter<<<(BB * SP + 255) / 256, 256, 0, stream>>>(out);
    k_marg<<<(VV + 255) / 256, 256, 0, stream>>>(out);
}